// GraphSAGEAutoencoder_77421080477950
// MI455X (gfx1250) — compile-verified
//
#include <hip/hip_runtime.h>
#include <hip/hip_bf16.h>

#define NODELEN 10
#define DEG     32
#define INF     20   // 2*NODELEN
#define H1      15
#define H2      10
#define EMBD    5
#define LDX     20   // LDS tile row stride (max layer width)
#define WAVES_PER_BLOCK 8

// packed-B chunk counts per layer: ceil(IN/4)
// L1(20):5  L2(15):4  L3(10):3  L4(5):2  L5(10):3  L6a(15):4  L6b(15):4 = 25
#define BP_O1  0
#define BP_O2  (BP_O1 + 5 * 64)
#define BP_O3  (BP_O2 + 4 * 64)
#define BP_O4  (BP_O3 + 3 * 64)
#define BP_O5  (BP_O4 + 2 * 64)
#define BP_O6A (BP_O5 + 3 * 64)
#define BP_O6B (BP_O6A + 4 * 64)
#define BP_TOT (BP_O6B + 4 * 64)   // 1600 floats

typedef float v2f __attribute__((ext_vector_type(2)));
typedef float v8f __attribute__((ext_vector_type(8)));

// ---------------------------------------------------------------------------
// Stage 1: h1[n,f] = mean_k features[adj[n,k], f]  ; h1[:,0] = 0
// One thread per (n,f); 32 fully-unrolled gathers against the L2-resident
// 8 MB feature table -> L2-BW bound, ~independent of HBM.
// ---------------------------------------------------------------------------
__global__ void hop1_kernel(const float* __restrict__ feat,
                            const int* __restrict__ adj,
                            float* __restrict__ h1, int N) {
    int id = blockIdx.x * blockDim.x + threadIdx.x;
    if (id >= N * NODELEN) return;
    int n = id / NODELEN, f = id % NODELEN;
    if (f == 0) { h1[id] = 0.0f; return; }
    const int* arow = adj + (size_t)n * DEG;
    float s = 0.0f;
#pragma unroll
    for (int k = 0; k < DEG; ++k) {
        int j = arow[k];
        s += feat[(size_t)j * NODELEN + f];
    }
    h1[id] = s * (1.0f / 32.0f);
}

// ---------------------------------------------------------------------------
// Stage 2: emb[n,0:10] = features[n] (col0 zeroed);
//          emb[n,10:20] = mean_k h1[adj[n,k]] (col10 zeroed)
// ---------------------------------------------------------------------------
__global__ void hop2_emb_kernel(const float* __restrict__ feat,
                                const int* __restrict__ adj,
                                const float* __restrict__ h1,
                                float* __restrict__ emb, int N) {
    int id = blockIdx.x * blockDim.x + threadIdx.x;
    if (id >= N * INF) return;
    int n = id / INF, f = id % INF;
    float v;
    if (f == 0 || f == NODELEN) {
        v = 0.0f;
    } else if (f < NODELEN) {
        v = feat[(size_t)n * NODELEN + f];
    } else {
        int ff = f - NODELEN;
        const int* arow = adj + (size_t)n * DEG;
        float s = 0.0f;
#pragma unroll
        for (int k = 0; k < DEG; ++k)
            s += h1[(size_t)arow[k] * NODELEN + ff];
        v = s * (1.0f / 32.0f);
    }
    emb[id] = v;
}

// ---------------------------------------------------------------------------
// Cooperative fill of a layer's weights into LDS in WMMA B-register packing:
//   Bp[c*64 + lane*2 + p] = B[4c + 2*(lane>=16) + p][nbase + (lane&15)]
// where B[k][n] = W[n][k] (torch layout), zero-padded beyond IN/OUT.
// Bias padded to 16 columns.
// ---------------------------------------------------------------------------
__device__ __forceinline__ void fill_B(float* __restrict__ Bp,
                                       float* __restrict__ bp,
                                       const float* __restrict__ W,
                                       const float* __restrict__ b,
                                       int IN, int OUT, int nbase) {
    const int C = (IN + 3) >> 2;
    for (int e = threadIdx.x; e < C * 64; e += blockDim.x) {
        int c = e >> 6;
        int r = (e >> 1) & 31;
        int p = e & 1;
        int k = 4 * c + 2 * (r >> 4) + p;
        int n = nbase + (r & 15);
        Bp[e] = (k < IN && n < OUT) ? W[n * IN + k] : 0.0f;
    }
    for (int e = threadIdx.x; e < 16; e += blockDim.x) {
        int n = nbase + e;
        bp[e] = (n < OUT) ? b[n] : 0.0f;
    }
}

// ---------------------------------------------------------------------------
// One MLP layer on a 16-node tile. All operands come from LDS, unconditional:
// A pair is contiguous in the stride-20 X tile (ds_load_b64), B pair is
// contiguous in the packed weight buffer (ds_load_b64). Padded columns of X
// are exact zeros, so chunk-rounded K reads are safe.
// ---------------------------------------------------------------------------
__device__ __forceinline__ v8f mlp_layer(const float* __restrict__ Bp,
                                         const float* __restrict__ bp,
                                         const float* X, int IN, bool relu) {
    const int lane = threadIdx.x & 31;
    const int half = lane >> 4;
    const int idx  = lane & 15;
    const int C = (IN + 3) >> 2;
    v8f d = {};
#pragma unroll
    for (int c = 0; c < C; ++c) {
        const int ka = 4 * c + 2 * half;
        v2f a, b;
        a.x = X[idx * LDX + ka];
        a.y = X[idx * LDX + ka + 1];
        b.x = Bp[c * 64 + lane * 2];
        b.y = Bp[c * 64 + lane * 2 + 1];
        d = __builtin_amdgcn_wmma_f32_16x16x4_f32(false, a, false, b,
                                                  (short)0, d, false, false);
    }
    const float bn = bp[idx];
#pragma unroll
    for (int r = 0; r < 8; ++r) {
        float v = d[r] + bn;
        d[r] = relu ? fmaxf(v, 0.0f) : v;
    }
    return d;
}

// Scatter a D-layout 16x16 tile back into the wave-private stride-20 X tile.
__device__ __forceinline__ void store_tile_lds(float* X, v8f d) {
    const int lane = threadIdx.x & 31;
    const int half = lane >> 4;
    const int idx  = lane & 15;
#pragma unroll
    for (int r = 0; r < 8; ++r)
        X[(r + 8 * half) * LDX + idx] = d[r];
}

// ---------------------------------------------------------------------------
// Stage 3: 6-layer autoencoder MLP, one wave per 16-node tile, f32 WMMA.
// ---------------------------------------------------------------------------
__global__ void __launch_bounds__(WAVES_PER_BLOCK * 32)
mlp_kernel(const float* __restrict__ emb,
           const float* __restrict__ We1, const float* __restrict__ be1,
           const float* __restrict__ We2, const float* __restrict__ be2,
           const float* __restrict__ We3, const float* __restrict__ be3,
           const float* __restrict__ Wd1, const float* __restrict__ bd1,
           const float* __restrict__ Wd2, const float* __restrict__ bd2,
           const float* __restrict__ Wd3, const float* __restrict__ bd3,
           float* __restrict__ enc, float* __restrict__ dec,
           int N, int ntiles) {
    __shared__ __align__(16) float tiles[WAVES_PER_BLOCK][16 * LDX];
    __shared__ __align__(16) float Bp[BP_TOT];
    __shared__ float Bias[7 * 16];

    // Block-wide one-time weight staging (all waves participate, then barrier).
    fill_B(Bp + BP_O1,  Bias + 0 * 16, We1, be1, INF,  H1,   0);
    fill_B(Bp + BP_O2,  Bias + 1 * 16, We2, be2, H1,   H2,   0);
    fill_B(Bp + BP_O3,  Bias + 2 * 16, We3, be3, H2,   EMBD, 0);
    fill_B(Bp + BP_O4,  Bias + 3 * 16, Wd1, bd1, EMBD, H2,   0);
    fill_B(Bp + BP_O5,  Bias + 4 * 16, Wd2, bd2, H2,   H1,   0);
    fill_B(Bp + BP_O6A, Bias + 5 * 16, Wd3, bd3, H1,   INF,  0);
    fill_B(Bp + BP_O6B, Bias + 6 * 16, Wd3, bd3, H1,   INF,  16);
    __syncthreads();

    const int wave = threadIdx.x >> 5;
    const int lane = threadIdx.x & 31;
    const int tile = blockIdx.x * WAVES_PER_BLOCK + wave;
    if (tile >= ntiles) return;           // wave-uniform; EXEC full for WMMA

    float* X = &tiles[wave][0];
    const int n0 = tile * 16;

    // Coalesced load of the contiguous 16x20 emb tile.
    const float* esrc = emb + (size_t)n0 * INF;
    for (int e = lane; e < 16 * INF; e += 32) {
        int row = n0 + e / INF;
        X[e] = (row < N) ? esrc[e] : 0.0f;
    }

    const int half = lane >> 4;
    const int idx  = lane & 15;

    v8f d;
    d = mlp_layer(Bp + BP_O1, Bias + 0 * 16, X, INF,  true);  store_tile_lds(X, d);
    d = mlp_layer(Bp + BP_O2, Bias + 1 * 16, X, H1,   true);  store_tile_lds(X, d);
    d = mlp_layer(Bp + BP_O3, Bias + 2 * 16, X, H2,   false); // encoded
    if (idx < EMBD) {
#pragma unroll
        for (int r = 0; r < 8; ++r) {
            int row = n0 + r + 8 * half;
            if (row < N) enc[(size_t)row * EMBD + idx] = d[r];
        }
    }
    store_tile_lds(X, d);   // padded cols idx>=5 are exact zeros

    d = mlp_layer(Bp + BP_O4, Bias + 3 * 16, X, EMBD, true);  store_tile_lds(X, d);
    d = mlp_layer(Bp + BP_O5, Bias + 4 * 16, X, H2,   true);  store_tile_lds(X, d);

    // decoded: OUT=20 -> two N-tiles
    v8f d0 = mlp_layer(Bp + BP_O6A, Bias + 5 * 16, X, H1, false);
    v8f d1 = mlp_layer(Bp + BP_O6B, Bias + 6 * 16, X, H1, false);
#pragma unroll
    for (int r = 0; r < 8; ++r) {
        int row = n0 + r + 8 * half;
        if (row < N) dec[(size_t)row * INF + idx] = d0[r];
    }
    if (idx < 4) {
#pragma unroll
        for (int r = 0; r < 8; ++r) {
            int row = n0 + r + 8 * half;
            if (row < N) dec[(size_t)row * INF + 16 + idx] = d1[r];
        }
    }
}

extern "C" void kernel_launch(void* const* d_in, const int* in_sizes, int n_in,
                              void* d_out, int out_size, void* d_ws, size_t ws_size,
                              hipStream_t stream) {
    const float* feat = (const float*)d_in[0];
    const int*   adj  = (const int*)d_in[1];
    const float* We1  = (const float*)d_in[2];
    const float* be1  = (const float*)d_in[3];
    const float* We2  = (const float*)d_in[4];
    const float* be2  = (const float*)d_in[5];
    const float* We3  = (const float*)d_in[6];
    const float* be3  = (const float*)d_in[7];
    const float* Wd1  = (const float*)d_in[8];
    const float* bd1  = (const float*)d_in[9];
    const float* Wd2  = (const float*)d_in[10];
    const float* bd2  = (const float*)d_in[11];
    const float* Wd3  = (const float*)d_in[12];
    const float* bd3  = (const float*)d_in[13];

    const int N = in_sizes[0] / NODELEN;

    float* out = (float*)d_out;
    float* enc = out;                            // [N,5]
    float* dec = out + (size_t)N * EMBD;         // [N,20]
    float* emb = out + (size_t)N * (EMBD + INF); // [N,20]
    float* h1  = (float*)d_ws;                   // [N,10] scratch

    const int t1 = N * NODELEN;
    hop1_kernel<<<(t1 + 255) / 256, 256, 0, stream>>>(feat, adj, h1, N);

    const int t2 = N * INF;
    hop2_emb_kernel<<<(t2 + 255) / 256, 256, 0, stream>>>(feat, adj, h1, emb, N);

    const int ntiles = (N + 15) / 16;
    const int blocks = (ntiles + WAVES_PER_BLOCK - 1) / WAVES_PER_BLOCK;
    mlp_kernel<<<blocks, WAVES_PER_BLOCK * 32, 0, stream>>>(
        emb, We1, be1, We2, be2, We3, be3, Wd1, bd1, Wd2, bd2, Wd3, bd3,
        enc, dec, N, ntiles);
}